// CTCLoss_64295660421952
// MI455X (gfx1250) — compile-verified
//
#include <hip/hip_runtime.h>
#include <cstdint>

// CTC loss forward on MI455X (gfx1250).
// One workgroup per batch element; S=2L+1 lattice states, one per thread
// (288 threads = 9 wave32). Sequential T-scan: alpha ping-pongs in LDS with
// ONE barrier per step. Emission rows stream global->LDS via
// global_load_async_to_lds_b128: every wave redundantly DMAs the row (same
// bytes, benign race; 8/9 requests hit WGP$/L2) so each wave gates only on
// its OWN ASYNCcnt -- no second barrier to publish the row. 8-slot ring
// prefetches 7 steps ahead to cover HBM latency.

#define NEGF (-1.0e30f)

__device__ __forceinline__ float lse3(float a, float b, float c) {
    float m = fmaxf(a, fmaxf(b, c));
    float r = __expf(a - m) + __expf(b - m) + __expf(c - m);
    return m + __logf(r);
}

// C fixed at 128 (one 512B row == 32 lanes x b128 per wave).
__global__ __launch_bounds__(288) void ctc_alpha_kernel(
    const float* __restrict__ pred,     // (B, T, 128) log-probs
    const int*   __restrict__ targets,  // (B, L)
    const int*   __restrict__ plen_arr, // (B,)
    const int*   __restrict__ tlen_arr, // (B,)
    float*       __restrict__ per_ex,   // (B,) loss_b / max(tl,1)
    int T, int L)
{
    constexpr int C     = 128;
    constexpr int DEPTH = 8;            // row ring slots (prefetch distance 7)
    const int b    = blockIdx.x;
    const int S    = 2 * L + 1;         // <= 258 supported
    const int tid  = threadIdx.x;
    const int lane = tid & 31;

    __shared__ __align__(16) float rowbuf[DEPTH][C]; // emission-row ring
    __shared__ float abuf[2][260];                   // alpha ping-pong, pads [0],[1]

    const float* predB = pred + (size_t)b * T * C;
    const int plen = min(plen_arr[b], T);
    const int tlen = tlen_arr[b];

    // Per-thread lattice-state constants.
    int   eclass = 0;      // blank = 0 for even states
    float skipm  = 0.0f;   // 1.0 iff (s-2 -> s) allowed
    if (tid < S) {
        if (tid & 1) {
            eclass = targets[(size_t)b * L + (tid >> 1)];
            if (tid >= 3) {
                int prev = targets[(size_t)b * L + (tid >> 1) - 1];
                skipm = (eclass != prev) ? 1.0f : 0.0f;
            }
            if (eclass == 0) skipm = 0.0f;
        }
        abuf[0][2 + tid] = NEGF;
        abuf[1][2 + tid] = NEGF;
    }
    if (tid < 2) { abuf[0][tid] = NEGF; abuf[1][tid] = NEGF; }

    const unsigned ldsrow0 = (unsigned)(uintptr_t)(&rowbuf[0][0]);
    const uint64_t gbase   = (uint64_t)(uintptr_t)predB;

    // Prime ring: every wave issues rows 0..6 for itself, then drains.
    for (int r = 0; r < DEPTH - 1; ++r) {
        if (r < T) {
            unsigned loff = ldsrow0 + (unsigned)(r * (C * 4) + lane * 16);
            unsigned goff = (unsigned)(r * (C * 4) + lane * 16);
            asm volatile("global_load_async_to_lds_b128 %0, %1, %2"
                         :: "v"(loff), "v"(goff), "s"(gbase) : "memory");
        }
    }
    asm volatile("s_wait_asynccnt 0x0" ::: "memory");

    // alpha_0: states 0 and 1 live; own value kept in a register.
    float myA = NEGF;
    if (tid == 0) myA = rowbuf[0][0];
    if (tid == 1) myA = rowbuf[0][eclass];
    if (tid < S)  abuf[0][2 + tid] = myA;
    __syncthreads();

    // Issue row 7 (end of "iteration 0").
    if (DEPTH - 1 < plen) {
        unsigned loff = ldsrow0 + (unsigned)(((DEPTH - 1) & (DEPTH - 1)) * (C * 4) + lane * 16);
        unsigned goff = (unsigned)((DEPTH - 1) * (C * 4) + lane * 16);
        asm volatile("global_load_async_to_lds_b128 %0, %1, %2"
                     :: "v"(loff), "v"(goff), "s"(gbase) : "memory");
    }

    int p = 0;
    for (int t = 1; t < plen; ++t) {
        // Gate on this wave's own async ops: with rows t..t+6 outstanding and
        // in-order completion, <=6 left  =>  row t resident. Near the tail
        // (<7 rows issued ahead) drain fully -- those ops are long in flight.
        if (t + DEPTH - 1 <= plen)
            asm volatile("s_wait_asynccnt 0x6" ::: "memory");
        else
            asm volatile("s_wait_asynccnt 0x0" ::: "memory");

        const float* row = rowbuf[t & (DEPTH - 1)];
        if (tid < S) {
            float a2 = abuf[p][1 + tid];
            float a3 = (skipm > 0.0f) ? abuf[p][tid] : NEGF;
            myA = lse3(myA, a2, a3) + row[eclass];
            abuf[p ^ 1][2 + tid] = myA;
        }
        __syncthreads();  // alpha exchange; also orders slot reuse below

        // Refill slot (t+7)&7 == (t-1)&7: its last readers finished before
        // barrier t-1 < barrier t, so the overwrite is safe.
        int r = t + DEPTH - 1;
        if (r < plen) {
            unsigned loff = ldsrow0 + (unsigned)((r & (DEPTH - 1)) * (C * 4) + lane * 16);
            unsigned goff = (unsigned)(r * (C * 4) + lane * 16);
            asm volatile("global_load_async_to_lds_b128 %0, %1, %2"
                         :: "v"(loff), "v"(goff), "s"(gbase) : "memory");
        }
        p ^= 1;
    }

    if (tid == 0) {
        int tl = max(tlen, 0);
        float l1 = (2 * tl - 1 >= 0 && 2 * tl - 1 < S) ? abuf[p][2 + 2 * tl - 1] : NEGF;
        float l2 = (2 * tl < S)                        ? abuf[p][2 + 2 * tl]     : NEGF;
        float m    = fmaxf(l1, l2);
        float loss = -(m + __logf(__expf(l1 - m) + __expf(l2 - m)));
        if (loss > 1e29f) loss = 0.0f;                 // zero_infinity
        per_ex[b] = loss / (float)max(tlen, 1);
    }
}

// Deterministic fixed-order mean over B partial losses.
__global__ void ctc_reduce_kernel(const float* __restrict__ per_ex,
                                  float* __restrict__ out, int B)
{
    if (blockIdx.x == 0 && threadIdx.x == 0) {
        float s = 0.0f;
        for (int i = 0; i < B; ++i) s += per_ex[i];
        out[0] = s / (float)B;
    }
}

extern "C" void kernel_launch(void* const* d_in, const int* in_sizes, int n_in,
                              void* d_out, int out_size, void* d_ws, size_t ws_size,
                              hipStream_t stream) {
    (void)n_in; (void)out_size; (void)ws_size;
    const float* pred    = (const float*)d_in[0];
    const int*   targets = (const int*)d_in[1];
    const int*   plen    = (const int*)d_in[2];
    const int*   tlen    = (const int*)d_in[3];

    const int B = in_sizes[2];                 // pred_lengths is (B,)
    const int L = in_sizes[1] / B;             // targets is (B, L)
    const int C = 128;                         // reference: 128 classes
    const int T = in_sizes[0] / (B * C);       // predictions is (B, T, C)

    float* per_ex = (float*)d_ws;              // B floats of scratch
    float* out    = (float*)d_out;

    ctc_alpha_kernel<<<B, 288, 0, stream>>>(pred, targets, plen, tlen,
                                            per_ex, T, L);
    ctc_reduce_kernel<<<1, 32, 0, stream>>>(per_ex, out, B);
}